// GCNProtein_62380105008026
// MI455X (gfx1250) — compile-verified
//
#include <hip/hip_runtime.h>
#include <hip/hip_bf16.h>

// GCN 2-layer on 1M nodes / 16M edges, feature widths 1 -> 3 -> 1.
//
// MI455X reasoning:
//  * All random-access targets (deg/agg1 = 4 MB, padded h1/agg2 = 16 MB) fit
//    in the 192 MB L2, so gathers and global_atomic_add_f32 resolve in L2.
//  * Edge index streams (~320 MB total across passes) come from HBM at
//    23.3 TB/s -> the kernel set is L2-atomic bound, roughly tens of us.
//  * Hidden vectors are padded to 4 floats so the layer-2 gather is one
//    GLOBAL_LOAD_B128 per edge.
//  * The final [N,3]@[3,1] projection is done with V_WMMA_F32_16X16X4_F32
//    (16 nodes per wave; B = W2 broadcast across all N columns so every
//    output column equals the desired dot product -> layout-robust).

#define NN 1000000
#define NE 16000000

typedef __attribute__((ext_vector_type(2))) float v2f;
typedef __attribute__((ext_vector_type(8))) float v8f;

__global__ void zero_f32(float* __restrict__ p, int n) {
  int i = blockIdx.x * blockDim.x + threadIdx.x;
  if (i < n) p[i] = 0.0f;
}

// deg[d] += 1 for each edge (segment_sum of ones over dst)
__global__ void deg_count(const int* __restrict__ dst, float* __restrict__ deg) {
  int e = blockIdx.x * blockDim.x + threadIdx.x;
  if (e < NE) atomicAdd(&deg[dst[e]], 1.0f);
}

// norm = rsqrt(max(deg,1));  h0 = feat * norm   (layer-1 pre-normalize)
__global__ void norm_h0(const float* __restrict__ feat, const float* __restrict__ deg,
                        float* __restrict__ norm, float* __restrict__ h0) {
  int i = blockIdx.x * blockDim.x + threadIdx.x;
  if (i < NN) {
    float nm = rsqrtf(fmaxf(deg[i], 1.0f));
    norm[i] = nm;
    h0[i] = feat[i] * nm;
  }
}

// agg1[dst] += h0[src]   (F=1 message passing)
__global__ void scatter1(const int* __restrict__ src, const int* __restrict__ dst,
                         const float* __restrict__ h0, float* __restrict__ agg1) {
  int e = blockIdx.x * blockDim.x + threadIdx.x;
  if (e < NE) atomicAdd(&agg1[dst[e]], h0[src[e]]);
}

// h1[n][k] = relu(agg1[n]*norm[n]*W1[k] + b1[k]) * norm[n]  (fold layer-2
// pre-normalize into the store). Padded to [N][4] for float4 gathers.
__global__ void layer1(const float* __restrict__ agg1, const float* __restrict__ norm,
                       const float* __restrict__ W1, const float* __restrict__ b1,
                       float* __restrict__ h1) {
  int i = blockIdx.x * blockDim.x + threadIdx.x;
  if (i < NN) {
    float nm = norm[i];
    float a = agg1[i] * nm;
    float4 o;
    o.x = fmaxf(fmaf(a, W1[0], b1[0]), 0.0f) * nm;
    o.y = fmaxf(fmaf(a, W1[1], b1[1]), 0.0f) * nm;
    o.z = fmaxf(fmaf(a, W1[2], b1[2]), 0.0f) * nm;
    o.w = 0.0f;
    reinterpret_cast<float4*>(h1)[i] = o;
  }
}

// agg2[dst][0..2] += h1[src][0..2]   (one B128 gather + 3 f32 atomics per edge)
__global__ void scatter2(const int* __restrict__ src, const int* __restrict__ dst,
                         const float* __restrict__ h1, float* __restrict__ agg2) {
  int e = blockIdx.x * blockDim.x + threadIdx.x;
  if (e < NE) {
    float4 v = reinterpret_cast<const float4*>(h1)[src[e]];
    float* p = agg2 + (size_t)dst[e] * 4;
    atomicAdd(p + 0, v.x);
    atomicAdd(p + 1, v.y);
    atomicAdd(p + 2, v.z);
  }
}

// out[n] = relu( (agg2[n][:]*norm[n]) . W2 + b2 ) via V_WMMA_F32_16X16X4_F32.
// 128 threads = 4 waves, 16 nodes per wave, 64 nodes per block; grid exactly
// covers N so EXEC is all ones at every WMMA (ISA requirement).
__global__ void layer2_wmma(const float* __restrict__ agg2, const float* __restrict__ norm,
                            const float* __restrict__ W2, const float* __restrict__ b2,
                            float* __restrict__ out) {
  int lane  = threadIdx.x & 31;
  int wave  = threadIdx.x >> 5;
  int base  = (blockIdx.x * 4 + wave) * 16;
  int m     = lane & 15;        // node row within tile
  int khalf = lane >> 4;        // 0 -> holds K=0,1 ; 1 -> holds K=2,3
  int node  = base + m;

  float nm = norm[node];
  // A 16x4 layout (ISA 7.12.2): lane = m + 16*(k>=2), VGPR = k&1 -> float2 load
  const float* ap = agg2 + (size_t)node * 4 + khalf * 2;
  v2f a;
  a.x = ap[0] * nm;
  a.y = ap[1] * nm;

  // B[k][n] = W2[k] for every n (k=3 column is zero): every D column equals
  // the desired dot product, independent of exact B lane mapping.
  v2f b;
  b.x = khalf ? W2[2] : W2[0];
  b.y = khalf ? 0.0f  : W2[1];

  v8f c = {};
  c = __builtin_amdgcn_wmma_f32_16x16x4_f32(
      /*neg_a=*/false, a, /*neg_b=*/false, b,
      /*c_mod=*/(short)0, c, /*reuse_a=*/false, /*reuse_b=*/false);

  // C/D layout: lanes 0-15 hold M=0..7 (VGPR r -> M=r), lanes 16-31 hold M=8..15.
  float bias = b2[0];
  if (lane == 0 || lane == 16) {
    int mb = base + (lane >> 4) * 8;
#pragma unroll
    for (int r = 0; r < 8; ++r)
      out[mb + r] = fmaxf(c[r] + bias, 0.0f);
  }
}

extern "C" void kernel_launch(void* const* d_in, const int* in_sizes, int n_in,
                              void* d_out, int out_size, void* d_ws, size_t ws_size,
                              hipStream_t stream) {
  const float* feat = (const float*)d_in[0];
  const float* W1   = (const float*)d_in[1];
  const float* b1   = (const float*)d_in[2];
  const float* W2   = (const float*)d_in[3];
  const float* b2   = (const float*)d_in[4];
  const int*   src  = (const int*)d_in[5];
  const int*   dst  = (const int*)d_in[6];
  float* out = (float*)d_out;

  float* ws = (float*)d_ws;
  // Workspace layout (floats): total 12*NN = 48 MB
  float* norm = ws;                       // [NN]
  float* h0   = ws + 1 * (size_t)NN;      // [NN]
  float* agg1 = ws + 2 * (size_t)NN;      // [NN]   (contiguous with deg)
  float* deg  = ws + 3 * (size_t)NN;      // [NN]
  float* h1   = ws + 4 * (size_t)NN;      // [NN][4] padded
  float* agg2 = ws + 8 * (size_t)NN;      // [NN][4] padded

  const int TB = 256;
  const int EDGE_BLOCKS = NE / TB;        // 62500 exact
  const int NODE_BLOCKS = (NN + TB - 1) / TB;

  // zero agg1+deg (2N contiguous) and agg2 (4N)
  zero_f32<<<(2 * NN) / TB, TB, 0, stream>>>(agg1, 2 * NN);
  zero_f32<<<(4 * NN) / TB, TB, 0, stream>>>(agg2, 4 * NN);

  deg_count<<<EDGE_BLOCKS, TB, 0, stream>>>(dst, deg);
  norm_h0<<<NODE_BLOCKS, TB, 0, stream>>>(feat, deg, norm, h0);
  scatter1<<<EDGE_BLOCKS, TB, 0, stream>>>(src, dst, h0, agg1);
  layer1<<<NODE_BLOCKS, TB, 0, stream>>>(agg1, norm, W1, b1, h1);
  scatter2<<<EDGE_BLOCKS, TB, 0, stream>>>(src, dst, h1, agg2);
  layer2_wmma<<<NN / 64, 128, 0, stream>>>(agg2, norm, W2, b2, out); // 15625 blocks
}